// ClusterLoss_46720654246269
// MI455X (gfx1250) — compile-verified
//
#include <hip/hip_runtime.h>
#include <hip/hip_bf16.h>
#include <math.h>

#define AS1 __attribute__((address_space(1)))
#define AS3 __attribute__((address_space(3)))

typedef int v4i __attribute__((vector_size(4 * sizeof(int))));

#ifndef __has_builtin
#define __has_builtin(x) 0
#endif

// ---------- CDNA5 async Global->LDS copy (ASYNCcnt-tracked) ----------
__device__ __forceinline__ void cl_async_load16(const float* g, float* l) {
#if __has_builtin(__builtin_amdgcn_global_load_async_to_lds_b128)
  __builtin_amdgcn_global_load_async_to_lds_b128(
      (AS1 v4i*)(void*)(const_cast<float*>(g)),
      (AS3 v4i*)(void*)l, 0, 0);
#else
  unsigned loff = (unsigned)(unsigned long long)(AS3 void*)l;
  asm volatile("global_load_async_to_lds_b128 %0, %1, off"
               :: "v"(loff), "v"(g) : "memory");
#endif
}

__device__ __forceinline__ void cl_wait_async_le1() {
#if __has_builtin(__builtin_amdgcn_s_wait_asynccnt)
  __builtin_amdgcn_s_wait_asynccnt(1);
#else
  asm volatile("s_wait_asynccnt 0x1" ::: "memory");
#endif
}

__device__ __forceinline__ void cl_wait_async_le0() {
#if __has_builtin(__builtin_amdgcn_s_wait_asynccnt)
  __builtin_amdgcn_s_wait_asynccnt(0);
#else
  asm volatile("s_wait_asynccnt 0x0" ::: "memory");
#endif
}

// ---------- small helpers ----------
__device__ __forceinline__ void cl_ins3(float v, int i,
                                        float& v0, int& i0,
                                        float& v1, int& i1,
                                        float& v2, int& i2) {
  if (v < v0)      { v2 = v1; i2 = i1; v1 = v0; i1 = i0; v0 = v; i0 = i; }
  else if (v < v1) { v2 = v1; i2 = i1; v1 = v;  i1 = i; }
  else if (v < v2) { v2 = v;  i2 = i; }
}

#define CL_TPB 256
#define CL_N   10000
#define CL_D   256

// ---------- kernel 0: zero accumulators ----------
__global__ void cl_zero(float* ws) {
  if (threadIdx.x < 4) ws[threadIdx.x] = 0.0f;
}

// ---------- kernel 1: mse + ||C||^2 + ||H||^2 (float4 grid-stride) ----------
__global__ void cl_elem(const float* __restrict__ X, const float* __restrict__ H,
                        const float* __restrict__ C, const float* __restrict__ M,
                        float* __restrict__ ws, int n4) {
  const int tid = threadIdx.x;
  int idx = blockIdx.x * blockDim.x + tid;
  const int stride = gridDim.x * blockDim.x;
  float mse = 0.f, c2 = 0.f, h2 = 0.f;
  const float4* X4 = (const float4*)X;
  const float4* H4 = (const float4*)H;
  const float4* C4 = (const float4*)C;
  const float4* M4 = (const float4*)M;
  for (int i = idx; i < n4; i += stride) {
    float4 x = X4[i], h = H4[i], c = C4[i], m = M4[i];
    float rx = (x.x - h.x + c.x) * m.x;
    float ry = (x.y - h.y + c.y) * m.y;
    float rz = (x.z - h.z + c.z) * m.z;
    float rw = (x.w - h.w + c.w) * m.w;
    mse += rx * rx + ry * ry + rz * rz + rw * rw;
    c2  += c.x * c.x + c.y * c.y + c.z * c.z + c.w * c.w;
    h2  += h.x * h.x + h.y * h.y + h.z * h.z + h.w * h.w;
  }
  __shared__ float red[CL_TPB][3];
  red[tid][0] = mse; red[tid][1] = c2; red[tid][2] = h2;
  __syncthreads();
  for (int s = CL_TPB / 2; s > 0; s >>= 1) {
    if (tid < s) {
      red[tid][0] += red[tid + s][0];
      red[tid][1] += red[tid + s][1];
      red[tid][2] += red[tid + s][2];
    }
    __syncthreads();
  }
  if (tid == 0) {
    atomicAdd(&ws[0], red[0][0]);
    atomicAdd(&ws[1], red[0][1]);
    atomicAdd(&ws[2], red[0][2]);
  }
}

// ---------- kernel 2: per-row top-3 (async-LDS streamed) + sim term ----------
__global__ void cl_topk_sim(const float* __restrict__ scores,
                            const int* __restrict__ mc_rows,
                            const float* __restrict__ H,
                            float* __restrict__ ws) {
  __shared__ float buf[2][CL_TPB * 4];  // double-buffered staging, 8 KB
  __shared__ float sv[CL_TPB][3];
  __shared__ int   si[CL_TPB][3];

  const int tid = threadIdx.x;
  const int r   = blockIdx.x;
  const int row = mc_rows[r];
  const float* src = scores + (size_t)row * CL_N;

  float v0 = INFINITY, v1 = INFINITY, v2 = INFINITY;
  int   i0 = 0, i1 = 0, i2 = 0;

  const int NV    = CL_N / 4;                    // 2500 float4 per row
  const int ITERS = (NV + CL_TPB - 1) / CL_TPB;  // 10

  // prologue: issue chunk 0
  {
    int v = tid;
    if (v < NV) cl_async_load16(src + (size_t)v * 4, &buf[0][tid * 4]);
  }
  for (int it = 0; it < ITERS; ++it) {
    if (it + 1 < ITERS) {
      int v = (it + 1) * CL_TPB + tid;
      if (v < NV) cl_async_load16(src + (size_t)v * 4, &buf[(it + 1) & 1][tid * 4]);
      cl_wait_async_le1();   // chunk `it` complete, chunk `it+1` may be in flight
    } else {
      cl_wait_async_le0();
    }
    int v = it * CL_TPB + tid;
    if (v < NV) {
      float4 f = *(const float4*)&buf[it & 1][tid * 4];
      int base = v * 4;
      cl_ins3(f.x, base + 0, v0, i0, v1, i1, v2, i2);
      cl_ins3(f.y, base + 1, v0, i0, v1, i1, v2, i2);
      cl_ins3(f.z, base + 2, v0, i0, v1, i1, v2, i2);
      cl_ins3(f.w, base + 3, v0, i0, v1, i1, v2, i2);
    }
  }

  // block-wide top-3 tree merge
  sv[tid][0] = v0; si[tid][0] = i0;
  sv[tid][1] = v1; si[tid][1] = i1;
  sv[tid][2] = v2; si[tid][2] = i2;
  __syncthreads();
  for (int s = CL_TPB / 2; s > 0; s >>= 1) {
    if (tid < s) {
      cl_ins3(sv[tid + s][0], si[tid + s][0], v0, i0, v1, i1, v2, i2);
      cl_ins3(sv[tid + s][1], si[tid + s][1], v0, i0, v1, i1, v2, i2);
      cl_ins3(sv[tid + s][2], si[tid + s][2], v0, i0, v1, i1, v2, i2);
      sv[tid][0] = v0; si[tid][0] = i0;
      sv[tid][1] = v1; si[tid][1] = i1;
      sv[tid][2] = v2; si[tid][2] = i2;
    }
    __syncthreads();
  }

  // broadcast final top-3 (t0 <= t1 <= t2)
  const float t0 = sv[0][0], t1 = sv[0][1], t2 = sv[0][2];
  const int   j0 = si[0][0], j1 = si[0][1], j2 = si[0][2];
  __syncthreads();  // everyone has read sv[0] before reuse

  // softmax(-t) with shift by max(-t) = -t0
  const float e0 = 1.0f;
  const float e1 = expf(t0 - t1);
  const float e2 = expf(t0 - t2);
  const float esum = e0 + e1 + e2;

  // ||H[row]-H[jk]||: D == blockDim, one dim per thread
  const float hr = H[(size_t)row * CL_D + tid];
  float d0 = hr - H[(size_t)j0 * CL_D + tid];
  float d1 = hr - H[(size_t)j1 * CL_D + tid];
  float d2 = hr - H[(size_t)j2 * CL_D + tid];
  sv[tid][0] = d0 * d0; sv[tid][1] = d1 * d1; sv[tid][2] = d2 * d2;
  __syncthreads();
  for (int s = CL_TPB / 2; s > 0; s >>= 1) {
    if (tid < s) {
      sv[tid][0] += sv[tid + s][0];
      sv[tid][1] += sv[tid + s][1];
      sv[tid][2] += sv[tid + s][2];
    }
    __syncthreads();
  }
  if (tid == 0) {
    float contrib = (e0 * sqrtf(sv[0][0]) +
                     e1 * sqrtf(sv[0][1]) +
                     e2 * sqrtf(sv[0][2])) / esum;
    atomicAdd(&ws[3], contrib);
  }
}

// ---------- kernel 3: combine ----------
__global__ void cl_final(const float* __restrict__ ws, float* __restrict__ out) {
  if (threadIdx.x == 0 && blockIdx.x == 0) {
    out[0] = ws[0] + ws[3] + 0.1f * sqrtf(ws[1]) + 0.01f * sqrtf(ws[2]);
  }
}

extern "C" void kernel_launch(void* const* d_in, const int* in_sizes, int n_in,
                              void* d_out, int out_size, void* d_ws, size_t ws_size,
                              hipStream_t stream) {
  // setup_inputs order: X, H, C, M, T, nM, row_scores, mc_rows
  const float* X          = (const float*)d_in[0];
  const float* H          = (const float*)d_in[1];
  const float* C          = (const float*)d_in[2];
  const float* M          = (const float*)d_in[3];
  const float* row_scores = (const float*)d_in[6];
  const int*   mc_rows    = (const int*)d_in[7];
  const int R = in_sizes[7];

  float* ws  = (float*)d_ws;
  float* out = (float*)d_out;

  const int n4 = (CL_N * CL_D) / 4;  // 640000 float4

  cl_zero<<<1, 32, 0, stream>>>(ws);
  cl_elem<<<640, CL_TPB, 0, stream>>>(X, H, C, M, ws, n4);
  cl_topk_sim<<<R, CL_TPB, 0, stream>>>(row_scores, mc_rows, H, ws);
  cl_final<<<1, 32, 0, stream>>>(ws, out);
}